// SACRSN_v44_23536420782649
// MI455X (gfx1250) — compile-verified
//
#include <hip/hip_runtime.h>
#include <hip/hip_bf16.h>
#include <math.h>

#define DIM    64
#define D2     128
#define B      16
#define SEQ    32
#define SLOTS  32
#define VOCAB  50257
#define MAXREC 8
#define TWO_PI_F 6.28318530717958647692f

#define NTILES  ((VOCAB + 15) / 16)      // 3142 column tiles
#define NBLK    ((NTILES + 7) / 8)       // 393 blocks, 8 waves/block

typedef float v2f __attribute__((ext_vector_type(2)));
typedef float v8f __attribute__((ext_vector_type(8)));

struct P {
  // inputs
  const int*   x_seq;
  const float *enc_emb, *norm_rg, *norm_rb, *norm_ig, *norm_ib;
  const float *q_Wr,*q_br,*q_Wi,*q_bi, *k_Wr,*k_br,*k_Wi,*k_bi, *v_Wr,*v_br,*v_Wi,*v_bi;
  const float *mem_gate_W,*mem_gate_b,*mem_addr_W,*mem_addr_b;
  const float *memnorm_rg,*memnorm_rb,*memnorm_ig,*memnorm_ib;
  const float *vq_emb, *arb_W, *arb_b, *dec_W, *dec_b;
  const float *halt_bias,*halt_scale,*input_gate;
  // workspace
  float *colnorm, *xr, *xi, *gwr, *gwi, *mr, *mi, *zf, *pang;
  unsigned long long* best;
  unsigned* counter;
  float *active,*avq,*apond,*aphase; int* fidx;
  float *scal;                       // [0]=ent_sum [1]=steps [2]=running
  float *outflat;                    // (B*SEQ, 128) decoder input
  float *stat_vq,*stat_pond,*stat_phase,*stat_ent;
  // outputs
  float *logits; float* stats_out; int* idxs_out;
};

__device__ __forceinline__ float sigm(float x){ return 1.f/(1.f+__expf(-x)); }
__device__ __forceinline__ float softplusf(float x){ return (x > 20.f) ? x : log1pf(__expf(x)); }

// -------------------- setup kernels --------------------

__global__ void sacrsn_colnorm(P p){
  int v = blockIdx.x*blockDim.x + threadIdx.x;
  if (v >= VOCAB) return;
  const float* e = p.vq_emb + (size_t)v*D2;
  float s=0.f;
  for (int k=0;k<D2;++k){ float x=e[k]; s += x*x; }
  p.colnorm[v] = s;
}

__global__ void sacrsn_init(P p){
  int stride = gridDim.x*blockDim.x;
  int g0 = blockIdx.x*blockDim.x+threadIdx.x;
  if (g0 == 0) *p.counter = 0u;
  for (int i = g0; i < B*SLOTS*DIM; i += stride){
    p.mr[i]=0.f; p.mi[i]=0.f;
    if (i < B*DIM){ p.gwr[i]=0.f; p.gwi[i]=0.f; }
  }
  for (int i = g0; i < SEQ*B*DIM; i += stride){
    int d = i % DIM; int b = (i/DIM) % B; int t = i/(DIM*B);
    int tok = p.x_seq[b*SEQ + t];
    p.xr[i] = p.enc_emb[(size_t)tok*D2 + d];
    p.xi[i] = p.enc_emb[(size_t)tok*D2 + DIM + d];
  }
}

// -------------------- boundary: seq_final(t_final) + seq_init(t_init) + ln_prep --------------------
__global__ void __launch_bounds__(256) sacrsn_boundary(P p, int t_final, int t_init){
  __shared__ float s_fr[B*DIM], s_fi[B*DIM];
  __shared__ float s_ww[B*SLOTS];
  __shared__ float s_wg[B];
  int tid=threadIdx.x;
  for (int i=tid;i<B*DIM;i+=256){ s_fr[i]=p.gwr[i]; s_fi[i]=p.gwi[i]; }
  __syncthreads();

  if (t_final >= 0){
    // ----- mem_write + outputs for step t_final -----
    if (tid < B){
      float g=p.mem_gate_b[0];
      for (int k=0;k<DIM;++k) g += s_fr[tid*DIM+k]*p.mem_gate_W[k] + s_fi[tid*DIM+k]*p.mem_gate_W[DIM+k];
      s_wg[tid]=sigm(g);
    }
    for (int i=tid;i<B*SLOTS;i+=256){
      int b=i/SLOTS, s=i%SLOTS;
      float a=p.mem_addr_b[s];
      for (int k=0;k<DIM;++k)
        a += s_fr[b*DIM+k]*p.mem_addr_W[k*SLOTS+s] + s_fi[b*DIM+k]*p.mem_addr_W[(DIM+k)*SLOTS+s];
      s_ww[i]=a;
    }
    __syncthreads();
    if (tid < B){
      float mx=-1e30f; for (int s=0;s<SLOTS;++s) mx=fmaxf(mx,s_ww[tid*SLOTS+s]);
      float sum=0.f;
      for (int s=0;s<SLOTS;++s){ float e=__expf(s_ww[tid*SLOTS+s]-mx); s_ww[tid*SLOTS+s]=e; sum+=e; }
      for (int s=0;s<SLOTS;++s) s_ww[tid*SLOTS+s] = s_ww[tid*SLOTS+s]/sum * s_wg[tid];  // eff
    }
    __syncthreads();
    for (int i=tid;i<B*SLOTS;i+=256){
      int b=i/SLOTS, s=i%SLOTS;
      float eff=s_ww[i];
      float* mrow=p.mr+(size_t)(b*SLOTS+s)*DIM;
      float sum=0.f, sq=0.f;
      for (int d=0;d<DIM;++d){ float x=(1.f-eff)*mrow[d]+eff*s_fr[b*DIM+d]; sum+=x; sq+=x*x; }
      float m=sum/DIM, var=sq/DIM-m*m, inv=rsqrtf(var+1e-5f);
      for (int d=0;d<DIM;++d){
        float x=(1.f-eff)*mrow[d]+eff*s_fr[b*DIM+d];
        mrow[d]=(x-m)*inv*p.memnorm_rg[d]+p.memnorm_rb[d];
      }
      float* mirow=p.mi+(size_t)(b*SLOTS+s)*DIM;
      sum=0.f; sq=0.f;
      for (int d=0;d<DIM;++d){ float x=(1.f-eff)*mirow[d]+eff*s_fi[b*DIM+d]; sum+=x; sq+=x*x; }
      m=sum/DIM; var=sq/DIM-m*m; inv=rsqrtf(var+1e-5f);
      for (int d=0;d<DIM;++d){
        float x=(1.f-eff)*mirow[d]+eff*s_fi[b*DIM+d];
        mirow[d]=(x-m)*inv*p.memnorm_ig[d]+p.memnorm_ib[d];
      }
    }
    for (int i=tid;i<B*DIM;i+=256){
      int b=i/DIM, d=i%DIM;
      p.outflat[((size_t)b*SEQ + t_final)*D2 + d]       = s_fr[i];
      p.outflat[((size_t)b*SEQ + t_final)*D2 + DIM + d] = s_fi[i];
    }
    if (tid < B) p.idxs_out[tid*SEQ + t_final] = p.fidx[tid];
    if (tid==0){
      float v=0.f, po=0.f, ph=0.f;
      for (int b=0;b<B;++b){ v+=p.avq[b]; po+=p.apond[b]; ph+=p.aphase[b]; }
      p.stat_vq[t_final]=v/B; p.stat_pond[t_final]=po/B; p.stat_phase[t_final]=ph/B;
      p.stat_ent[t_final]=p.scal[0]/fmaxf(p.scal[1],1.f);
    }
    __syncthreads();
  }

  if (t_init >= 0){
    // ----- input mixing + carry reset -----
    float alpha = sigm(p.input_gate[0]);
    for (int i=tid;i<B*DIM;i+=256){
      float gr = alpha*s_fr[i] + (1.f-alpha)*p.xr[t_init*B*DIM + i];
      float gi = alpha*s_fi[i] + (1.f-alpha)*p.xi[t_init*B*DIM + i];
      s_fr[i]=gr; s_fi[i]=gi;
      p.gwr[i]=gr; p.gwi[i]=gi;
      p.pang[i]=atan2f(gi, gr);
    }
    if (tid < B){
      p.active[tid]=1.f; p.avq[tid]=0.f; p.apond[tid]=0.f;
      p.aphase[tid]=0.f; p.fidx[tid]=0;
    }
    if (tid==0){ p.scal[0]=0.f; p.scal[1]=0.f; p.scal[2]=1.f; }
    __syncthreads();
    // ----- layernorm -> zf, reset argmin -----
    if (tid < B){
      int b=tid;
      float sum=0.f, sq=0.f;
      for (int d=0;d<DIM;++d){ float x=s_fr[b*DIM+d]; sum+=x; sq+=x*x; }
      float m=sum/DIM, var=sq/DIM-m*m, inv=rsqrtf(var+1e-5f);
      for (int d=0;d<DIM;++d) p.zf[b*D2+d] = (s_fr[b*DIM+d]-m)*inv*p.norm_rg[d]+p.norm_rb[d];
      sum=0.f; sq=0.f;
      for (int d=0;d<DIM;++d){ float x=s_fi[b*DIM+d]; sum+=x; sq+=x*x; }
      m=sum/DIM; var=sq/DIM-m*m; inv=rsqrtf(var+1e-5f);
      for (int d=0;d<DIM;++d) p.zf[b*D2+DIM+d] = (s_fi[b*DIM+d]-m)*inv*p.norm_ig[d]+p.norm_ib[d];
      p.best[b] = 0xFFFFFFFFFFFFFFFFULL;
    }
  }
}

// -------------------- fused phase: WMMA VQ scan + last-block recurrent update --------------------
__global__ void __launch_bounds__(256) sacrsn_phase(P p){
  __shared__ float s_zf[B*D2];
  __shared__ float s_zq[B*D2];
  __shared__ float s_qr[B*DIM], s_qi[B*DIM], s_kr[B*DIM], s_ki[B*DIM], s_vr[B*DIM], s_vi[B*DIM];
  __shared__ float s_rr[B*DIM], s_ri[B*DIM];
  __shared__ float s_sim[B*SLOTS];
  __shared__ float s_gate[B], s_g3[B*3], s_vql[B], s_tmp[B], s_diff[B], s_act[B];
  __shared__ int   s_idx[B];
  __shared__ float s_ent;
  __shared__ int   s_last;
  int tid = threadIdx.x;

  // stage the shared A-panel (zf) once per block
  for (int i=tid;i<B*D2;i+=256) s_zf[i]=p.zf[i];
  __syncthreads();

  // ---- WMMA codebook scan: S = zf(16x128) @ E^T(128x16), argmin(colnorm - 2S) ----
  {
    int lane = tid & 31;
    int wave = tid >> 5;
    int tile = blockIdx.x*8 + wave;
    if (tile >= NTILES) tile = NTILES - 1;          // clamp: EXEC stays full for WMMA
    int col  = tile*16 + (lane & 15);
    int colC = (col < VOCAB) ? col : (VOCAB - 1);
    int row  = lane & 15;
    int khalf = (lane < 16) ? 0 : 2;

    v8f acc = {};
    for (int k0 = 0; k0 < D2; k0 += 4){
      v2f a, b;
      a.x = s_zf[row*D2 + k0 + khalf];              // A from LDS (broadcast panel)
      a.y = s_zf[row*D2 + k0 + khalf + 1];
      b.x = p.vq_emb[(size_t)colC*D2 + k0 + khalf]; // E^T[k][col] (L2-resident)
      b.y = p.vq_emb[(size_t)colC*D2 + k0 + khalf + 1];
      acc = __builtin_amdgcn_wmma_f32_16x16x4_f32(false, a, false, b,
                                                  (short)0, acc, false, false);
    }
    float cn = p.colnorm[colC];
    int rbase = (lane < 16) ? 0 : 8;
    for (int r = 0; r < 8; ++r){
      float s = cn - 2.f*acc[r];
      int c = colC;
      for (int m = 1; m < 16; m <<= 1){
        float so = __shfl_xor(s, m, 32);
        int   co = __shfl_xor(c, m, 32);
        if (so < s || (so == s && co < c)){ s = so; c = co; }
      }
      if ((lane & 15) == 0){
        unsigned su = __float_as_uint(s);
        su = (su & 0x80000000u) ? ~su : (su | 0x80000000u);
        unsigned long long key = ((unsigned long long)su << 32) | (unsigned)c;
        atomicMin(&p.best[rbase + r], key);
      }
    }
  }
  __threadfence();                                   // release our atomicMin results
  __syncthreads();
  if (tid==0){
    unsigned old = atomicAdd(p.counter, 1u);
    s_last = (old == (unsigned)gridDim.x - 1u) ? 1 : 0;
  }
  __syncthreads();
  if (!s_last) return;

  // ================= last block: recurrent step update =================
  __threadfence();                                   // acquire all blocks' results
  if (tid==0) *p.counter = 0u;
  if (tid < B){
    s_idx[tid] = (int)(p.best[tid] & 0xFFFFFFFFull);
    s_act[tid] = p.active[tid];
    s_diff[tid]= 0.f;
  }
  __syncthreads();
  for (int i=tid;i<B*D2;i+=256){
    int b=i/D2, d=i%D2;
    s_zq[i] = p.vq_emb[(size_t)s_idx[b]*D2 + d];
  }
  __syncthreads();
  if (tid < B){
    float s=0.f;
    for (int d=0;d<D2;++d){ float df=s_zq[tid*D2+d]-s_zf[tid*D2+d]; s+=df*df; }
    s_vql[tid] = 1.25f * s / D2;                     // loss + 0.25*loss (ST forward)
    int c=0; for (int j=0;j<B;++j) c += (s_idx[j]==s_idx[tid]);
    float a=(float)c/(float)B;
    s_tmp[tid] = -(a*__logf(a+1e-10f))/(float)c;
  }
  __syncthreads();
  if (tid==0){ float e=0.f; for (int j=0;j<B;++j) e+=s_tmp[j]; s_ent=e; }

  // q/k/v complex linears (64x64 -> VALU; too small for WMMA tile overhead)
  for (int i=tid;i<B*DIM;i+=256){
    int b=i/DIM, d=i%DIM;
    const float* cr=&s_zf[b*D2]; const float* ci=cr+DIM;
    float qrr=0,qii=0,qir=0,qri=0, krr=0,kii=0,kir=0,kri=0, vrr=0,vii=0,vir=0,vri=0;
    for (int k=0;k<DIM;++k){
      float xr=cr[k], xi=ci[k];
      float a,bw;
      a=p.q_Wr[k*DIM+d]; bw=p.q_Wi[k*DIM+d]; qrr+=xr*a; qir+=xi*a; qii+=xi*bw; qri+=xr*bw;
      a=p.k_Wr[k*DIM+d]; bw=p.k_Wi[k*DIM+d]; krr+=xr*a; kir+=xi*a; kii+=xi*bw; kri+=xr*bw;
      a=p.v_Wr[k*DIM+d]; bw=p.v_Wi[k*DIM+d]; vrr+=xr*a; vir+=xi*a; vii+=xi*bw; vri+=xr*bw;
    }
    s_qr[i]=qrr+p.q_br[d]-(qii+p.q_bi[d]);  s_qi[i]=qir+p.q_br[d]+(qri+p.q_bi[d]);
    s_kr[i]=krr+p.k_br[d]-(kii+p.k_bi[d]);  s_ki[i]=kir+p.k_br[d]+(kri+p.k_bi[d]);
    s_vr[i]=vrr+p.v_br[d]-(vii+p.v_bi[d]);  s_vi[i]=vir+p.v_br[d]+(vri+p.v_bi[d]);
  }
  for (int i=tid;i<B*SLOTS;i+=256){
    int b=i/SLOTS, s=i%SLOTS;
    const float* mr=p.mr+(size_t)(b*SLOTS+s)*DIM;
    const float* mi=p.mi+(size_t)(b*SLOTS+s)*DIM;
    const float* cr=&s_zf[b*D2]; const float* ci=cr+DIM;
    float acc=0.f;
    for (int d=0;d<DIM;++d) acc += mr[d]*cr[d] + mi[d]*ci[d];
    s_sim[i]=acc;
  }
  __syncthreads();
  if (tid < B){
    float g=0.f;
    for (int d=0;d<DIM;++d) g += s_qr[tid*DIM+d]*s_kr[tid*DIM+d] + s_qi[tid*DIM+d]*s_ki[tid*DIM+d];
    s_gate[tid]=sigm(g);
    float mx=-1e30f; for (int s=0;s<SLOTS;++s) mx=fmaxf(mx,s_sim[tid*SLOTS+s]);
    float sum=0.f;
    for (int s=0;s<SLOTS;++s){ float e=__expf(s_sim[tid*SLOTS+s]-mx); s_sim[tid*SLOTS+s]=e; sum+=e; }
    for (int s=0;s<SLOTS;++s) s_sim[tid*SLOTS+s]/=sum;
    float l[3]; float amx=-1e30f;
    for (int j=0;j<3;++j){
      float a=p.arb_b[j];
      for (int k=0;k<D2;++k) a += s_zf[tid*D2+k]*p.arb_W[k*3+j];
      l[j]=a; amx=fmaxf(amx,a);
    }
    float asum=0.f;
    for (int j=0;j<3;++j){ l[j]=__expf(l[j]-amx); asum+=l[j]; }
    for (int j=0;j<3;++j) s_g3[tid*3+j]=l[j]/asum;
  }
  __syncthreads();
  for (int i=tid;i<B*DIM;i+=256){
    int b=i/DIM, d=i%DIM;
    float ar=0.f, ai=0.f;
    for (int s=0;s<SLOTS;++s){
      float w=s_sim[b*SLOTS+s];
      ar += w*p.mr[(size_t)(b*SLOTS+s)*DIM+d];
      ai += w*p.mi[(size_t)(b*SLOTS+s)*DIM+d];
    }
    s_rr[i]=ar; s_ri[i]=ai;
  }
  __syncthreads();
  // combine + angle + gw update; s_qr/s_qi reused as new-gw staging (gate already consumed)
  for (int i=tid;i<B*DIM;i+=256){
    int b=i/DIM, d=i%DIM;
    float cr=s_zf[b*D2+d], ci=s_zf[b*D2+DIM+d];
    float g0=s_g3[b*3], g1=s_g3[b*3+1], g2=s_g3[b*3+2];
    float gr=s_vr[i]*s_gate[b], gi=s_vi[i]*s_gate[b];
    float vr=s_zq[b*D2+d],  vi=s_zq[b*D2+DIM+d];
    float ur=g0*gr + g1*s_rr[i] + g2*vr;
    float ui=g0*gi + g1*s_ri[i] + g2*vi;
    float car=0.6f*cr + 0.4f*ur;
    float cai=0.6f*ci + 0.4f*ui;
    float ang=atan2f(cai, car);
    float df=fabsf(ang - p.pang[i]); df=fminf(df, TWO_PI_F - df);
    atomicAdd(&s_diff[b], df);
    p.pang[i]=ang;                                   // carry passes ang always
    float ogr=p.gwr[i], ogi=p.gwi[i];
    float ngr = (s_act[b] > 0.5f) ? car : ogr;
    float ngi = (s_act[b] > 0.5f) ? cai : ogi;
    s_qr[i]=ngr; s_qi[i]=ngi;
    p.gwr[i]=ngr; p.gwi[i]=ngi;
  }
  __syncthreads();
  if (tid < B){
    int b=tid;
    float am=s_act[b];
    p.aphase[b] += am*(s_diff[b]/(float)DIM);
    p.apond[b]  += am*0.01f;
    if (am > 0.5f){ p.avq[b]=s_vql[b]; p.fidx[b]=s_idx[b]; }
    float bias  = softplusf(p.halt_bias[0]);
    float scale = softplusf(p.halt_scale[0]);
    float stop  = (sigm(bias - scale*s_vql[b]) > 0.5f) ? 1.f : 0.f;
    float na = am*(1.f-stop);
    p.active[b]=na; s_act[b]=na;
    // ---- next phase's layernorm -> zf, reset argmin ----
    float sum=0.f, sq=0.f;
    for (int d=0;d<DIM;++d){ float x=s_qr[b*DIM+d]; sum+=x; sq+=x*x; }
    float m=sum/DIM, var=sq/DIM-m*m, inv=rsqrtf(var+1e-5f);
    for (int d=0;d<DIM;++d) p.zf[b*D2+d] = (s_qr[b*DIM+d]-m)*inv*p.norm_rg[d]+p.norm_rb[d];
    sum=0.f; sq=0.f;
    for (int d=0;d<DIM;++d){ float x=s_qi[b*DIM+d]; sum+=x; sq+=x*x; }
    m=sum/DIM; var=sq/DIM-m*m; inv=rsqrtf(var+1e-5f);
    for (int d=0;d<DIM;++d) p.zf[b*D2+DIM+d] = (s_qi[b*DIM+d]-m)*inv*p.norm_ig[d]+p.norm_ib[d];
    p.best[b] = 0xFFFFFFFFFFFFFFFFULL;
  }
  __syncthreads();
  if (tid==0){
    float running=p.scal[2];
    p.scal[0] += s_ent*running;
    p.scal[1] += running;
    float sa=0.f; for (int b=0;b<B;++b) sa += s_act[b];
    p.scal[2] = running * ((sa > 0.f) ? 1.f : 0.f);
  }
}

// -------------------- WMMA decoder: (512x128) @ (128x50257) --------------------
__global__ void __launch_bounds__(256) sacrsn_decoder(P p){
  int lane=threadIdx.x & 31;
  int wave=threadIdx.x >> 5;
  int tn=blockIdx.x*8 + wave;
  if (tn >= NTILES) tn = NTILES - 1;
  int tm=blockIdx.y;
  int arow = tm*16 + (lane & 15);
  int col  = tn*16 + (lane & 15);
  int colC = (col < VOCAB) ? col : (VOCAB - 1);
  int khalf=(lane<16)?0:2;
  v8f acc = {};
  for (int k0=0;k0<D2;k0+=4){
    v2f a,b;
    a.x = p.outflat[(size_t)arow*D2 + k0 + khalf];
    a.y = p.outflat[(size_t)arow*D2 + k0 + khalf + 1];
    b.x = p.dec_W[(size_t)(k0+khalf)*VOCAB + colC];
    b.y = p.dec_W[(size_t)(k0+khalf+1)*VOCAB + colC];
    acc = __builtin_amdgcn_wmma_f32_16x16x4_f32(false, a, false, b,
                                                (short)0, acc, false, false);
  }
  if (col < VOCAB){
    float bv=p.dec_b[col];
    int rb=(lane<16)?0:8;
    for (int r=0;r<8;++r){
      int orow = tm*16 + rb + r;
      p.logits[(size_t)orow*VOCAB + col] = acc[r] + bv;
    }
  }
}

__global__ void sacrsn_final_stats(P p){
  int tid=threadIdx.x;
  if (tid < 4){
    // jax pytree dict order (sorted keys): ent, phase, ponder, vq
    const float* src = (tid==0)?p.stat_ent : (tid==1)?p.stat_phase : (tid==2)?p.stat_pond : p.stat_vq;
    float s=0.f; for (int t=0;t<SEQ;++t) s+=src[t];
    p.stats_out[tid]=s/SEQ;
  }
}

// -------------------- host --------------------

extern "C" void kernel_launch(void* const* d_in, const int* in_sizes, int n_in,
                              void* d_out, int out_size, void* d_ws, size_t ws_size,
                              hipStream_t stream){
  (void)in_sizes; (void)n_in; (void)out_size; (void)ws_size;
  P p;
  p.x_seq      = (const int*)  d_in[0];
  p.enc_emb    = (const float*)d_in[1];
  p.norm_rg    = (const float*)d_in[2];  p.norm_rb = (const float*)d_in[3];
  p.norm_ig    = (const float*)d_in[4];  p.norm_ib = (const float*)d_in[5];
  p.q_Wr=(const float*)d_in[6];  p.q_br=(const float*)d_in[7];
  p.q_Wi=(const float*)d_in[8];  p.q_bi=(const float*)d_in[9];
  p.k_Wr=(const float*)d_in[10]; p.k_br=(const float*)d_in[11];
  p.k_Wi=(const float*)d_in[12]; p.k_bi=(const float*)d_in[13];
  p.v_Wr=(const float*)d_in[14]; p.v_br=(const float*)d_in[15];
  p.v_Wi=(const float*)d_in[16]; p.v_bi=(const float*)d_in[17];
  p.mem_gate_W=(const float*)d_in[18]; p.mem_gate_b=(const float*)d_in[19];
  p.mem_addr_W=(const float*)d_in[20]; p.mem_addr_b=(const float*)d_in[21];
  p.memnorm_rg=(const float*)d_in[22]; p.memnorm_rb=(const float*)d_in[23];
  p.memnorm_ig=(const float*)d_in[24]; p.memnorm_ib=(const float*)d_in[25];
  p.vq_emb=(const float*)d_in[26];
  p.arb_W=(const float*)d_in[27]; p.arb_b=(const float*)d_in[28];
  p.dec_W=(const float*)d_in[29]; p.dec_b=(const float*)d_in[30];
  p.halt_bias=(const float*)d_in[31]; p.halt_scale=(const float*)d_in[32];
  p.input_gate=(const float*)d_in[33];

  char* w = (char*)d_ws;
  auto take = [&](size_t bytes)->void* {
    void* r = (void*)w;
    w += (bytes + 255) & ~(size_t)255;
    return r;
  };
  p.colnorm = (float*)take(sizeof(float)*VOCAB);
  p.xr      = (float*)take(sizeof(float)*SEQ*B*DIM);
  p.xi      = (float*)take(sizeof(float)*SEQ*B*DIM);
  p.gwr     = (float*)take(sizeof(float)*B*DIM);
  p.gwi     = (float*)take(sizeof(float)*B*DIM);
  p.mr      = (float*)take(sizeof(float)*B*SLOTS*DIM);
  p.mi      = (float*)take(sizeof(float)*B*SLOTS*DIM);
  p.zf      = (float*)take(sizeof(float)*B*D2);
  p.pang    = (float*)take(sizeof(float)*B*DIM);
  p.best    = (unsigned long long*)take(sizeof(unsigned long long)*B);
  p.counter = (unsigned*)take(sizeof(unsigned));
  p.active  = (float*)take(sizeof(float)*B);
  p.avq     = (float*)take(sizeof(float)*B);
  p.apond   = (float*)take(sizeof(float)*B);
  p.aphase  = (float*)take(sizeof(float)*B);
  p.fidx    = (int*)  take(sizeof(int)*B);
  p.scal    = (float*)take(sizeof(float)*4);
  p.outflat = (float*)take(sizeof(float)*B*SEQ*D2);
  p.stat_vq   = (float*)take(sizeof(float)*SEQ);
  p.stat_pond = (float*)take(sizeof(float)*SEQ);
  p.stat_phase= (float*)take(sizeof(float)*SEQ);
  p.stat_ent  = (float*)take(sizeof(float)*SEQ);

  const size_t LOGITS_N = (size_t)B*SEQ*VOCAB;
  p.logits    = (float*)d_out;
  p.stats_out = p.logits + LOGITS_N;
  p.idxs_out  = (int*)(p.logits + LOGITS_N + 4);

  sacrsn_colnorm<<<(VOCAB+255)/256, 256, 0, stream>>>(p);
  sacrsn_init   <<<128, 256, 0, stream>>>(p);
  for (int t=0; t<SEQ; ++t){
    sacrsn_boundary<<<1, 256, 0, stream>>>(p, t-1, t);
    for (int r=0; r<MAXREC; ++r){
      sacrsn_phase<<<NBLK, 256, 0, stream>>>(p);
    }
  }
  sacrsn_boundary<<<1, 256, 0, stream>>>(p, SEQ-1, -1);
  dim3 dg(NBLK, (B*SEQ)/16);
  sacrsn_decoder    <<<dg, 256, 0, stream>>>(p);
  sacrsn_final_stats<<<1, 64, 0, stream>>>(p);
}